// SAGEConvolution_81140522156079
// MI455X (gfx1250) — compile-verified
//
#include <hip/hip_runtime.h>

#define N_NODES 100000
#define N_EDGES 1000000
#define F 64
#define NCLS 2

typedef __attribute__((ext_vector_type(2))) float v2f;
typedef __attribute__((ext_vector_type(8))) float v8f;

// ---------------------------------------------------------------- utilities

__global__ void zero_f32(float* __restrict__ p, int n) {
  int i = blockIdx.x * blockDim.x + threadIdx.x;
  int stride = gridDim.x * blockDim.x;
  for (; i < n; i += stride) p[i] = 0.0f;
}

__global__ void degree_count(const int* __restrict__ dst, float* __restrict__ cnt, int e) {
  int t = blockIdx.x * blockDim.x + threadIdx.x;
  if (t < e) atomicAdd(&cnt[dst[t]], 1.0f);
}

// one thread per (edge, feature); lanes cover consecutive features of one edge
// -> coalesced gather of feat[src] and coalesced global_atomic_add_f32 bursts.
__global__ void scatter_add(const float* __restrict__ feat, const int* __restrict__ src,
                            const int* __restrict__ dst, float* __restrict__ sum, int e) {
  int t = blockIdx.x * blockDim.x + threadIdx.x;
  if (t >= e * F) return;
  int edge = t >> 6;
  int f = t & (F - 1);
  atomicAdd(&sum[(size_t)dst[edge] * F + f], feat[(size_t)src[edge] * F + f]);
}

// ------------------------------------------------- layer 1: fused mean+GEMM
// h = relu( (sum/deg) @ W1l^T + x @ W1r^T + b1 ), all in f32 WMMA 16x16x4.
// Block = 128 threads = 4 waves; each wave produces a 16x64 output tile.

__global__ __launch_bounds__(128) void sage_layer1_wmma(
    const float* __restrict__ x, const float* __restrict__ sum,
    const float* __restrict__ cnt, const float* __restrict__ W1l,
    const float* __restrict__ W1r, const float* __restrict__ b1,
    float* __restrict__ h) {
  __shared__ __align__(16) float sW[2 * F * F];  // [0,4096): W1l  [4096,8192): W1r
  int tid = threadIdx.x;

  const float4* wl4 = (const float4*)W1l;
  const float4* wr4 = (const float4*)W1r;
  float4* s4 = (float4*)sW;
  for (int i = tid; i < (F * F) / 4; i += 128) {
    s4[i] = wl4[i];
    s4[i + (F * F) / 4] = wr4[i];
  }
  __syncthreads();

  int wave = tid >> 5;
  int lane = tid & 31;
  int half = lane >> 4;  // 0: K even pair base, 1: +2
  int r = lane & 15;
  int m0 = (blockIdx.x * 4 + wave) * 16;
  if (m0 >= N_NODES) return;  // wave-uniform: EXEC stays all-1s for WMMA

  int m = m0 + r;  // A-matrix row for this lane
  float inv = 1.0f / fmaxf(cnt[m], 1.0f);

  v8f acc[4];
  for (int nt = 0; nt < 4; ++nt) {
    float bb = b1[nt * 16 + r];
    for (int v = 0; v < 8; ++v) acc[nt][v] = bb;
  }

  const float* sumRow = sum + (size_t)m * F;
  const float* xRow = x + (size_t)m * F;

#pragma unroll
  for (int kt = 0; kt < 16; ++kt) {
    int kA = kt * 4 + 2 * half;
    v2f aA = *(const v2f*)(sumRow + kA);
    aA = aA * inv;  // fold mean divide into A fragment
    v2f aX = *(const v2f*)(xRow + kA);
#pragma unroll
    for (int nt = 0; nt < 4; ++nt) {
      int n = nt * 16 + r;
      v2f bL = *(const v2f*)&sW[n * F + kA];
      v2f bR = *(const v2f*)&sW[F * F + n * F + kA];
      acc[nt] = __builtin_amdgcn_wmma_f32_16x16x4_f32(
          false, aA, false, bL, (short)0, acc[nt], false, false);
      acc[nt] = __builtin_amdgcn_wmma_f32_16x16x4_f32(
          false, aX, false, bR, (short)0, acc[nt], false, false);
    }
  }

  // C/D layout: VGPR v, lane L -> row = v + 8*(L>>4), col = L&15
#pragma unroll
  for (int nt = 0; nt < 4; ++nt)
#pragma unroll
    for (int v = 0; v < 8; ++v) {
      int row = m0 + v + 8 * half;
      h[(size_t)row * F + nt * 16 + r] = fmaxf(acc[nt][v], 0.0f);
    }
}

// ------------------------------------------------- layer 2: fused mean+GEMM
// out = (sum2/deg) @ W2l^T + h @ W2r^T + b2 ; W2 (2x64) zero-padded to 16x64.

__global__ __launch_bounds__(128) void sage_layer2_wmma(
    const float* __restrict__ hf, const float* __restrict__ sum,
    const float* __restrict__ cnt, const float* __restrict__ W2l,
    const float* __restrict__ b2, const float* __restrict__ W2r,
    float* __restrict__ out) {
  __shared__ __align__(16) float sW[2 * 16 * F];  // padded W2l | padded W2r
  int tid = threadIdx.x;
  for (int i = tid; i < 16 * F; i += 128) {
    float vl = (i < NCLS * F) ? W2l[i] : 0.0f;
    float vr = (i < NCLS * F) ? W2r[i] : 0.0f;
    sW[i] = vl;
    sW[16 * F + i] = vr;
  }
  __syncthreads();

  int wave = tid >> 5;
  int lane = tid & 31;
  int half = lane >> 4;
  int r = lane & 15;
  int m0 = (blockIdx.x * 4 + wave) * 16;
  if (m0 >= N_NODES) return;

  int m = m0 + r;
  float inv = 1.0f / fmaxf(cnt[m], 1.0f);

  v8f acc;
  float bb = (r < NCLS) ? b2[r] : 0.0f;
  for (int v = 0; v < 8; ++v) acc[v] = bb;

  const float* sumRow = sum + (size_t)m * F;
  const float* hRow = hf + (size_t)m * F;

#pragma unroll
  for (int kt = 0; kt < 16; ++kt) {
    int kA = kt * 4 + 2 * half;
    v2f aA = *(const v2f*)(sumRow + kA);
    aA = aA * inv;
    v2f aH = *(const v2f*)(hRow + kA);
    v2f bL = *(const v2f*)&sW[r * F + kA];
    v2f bR = *(const v2f*)&sW[16 * F + r * F + kA];
    acc = __builtin_amdgcn_wmma_f32_16x16x4_f32(
        false, aA, false, bL, (short)0, acc, false, false);
    acc = __builtin_amdgcn_wmma_f32_16x16x4_f32(
        false, aH, false, bR, (short)0, acc, false, false);
  }

  if (r < NCLS) {  // divergence only AFTER all WMMAs
#pragma unroll
    for (int v = 0; v < 8; ++v) {
      int row = m0 + v + 8 * half;
      out[(size_t)row * NCLS + r] = acc[v];
    }
  }
}

// ---------------------------------------------------------------- launcher

extern "C" void kernel_launch(void* const* d_in, const int* in_sizes, int n_in,
                              void* d_out, int out_size, void* d_ws, size_t ws_size,
                              hipStream_t stream) {
  const float* x = (const float*)d_in[0];
  const int* ei = (const int*)d_in[1];
  const float* W1l = (const float*)d_in[2];
  const float* b1 = (const float*)d_in[3];
  const float* W1r = (const float*)d_in[4];
  const float* W2l = (const float*)d_in[5];
  const float* b2 = (const float*)d_in[6];
  const float* W2r = (const float*)d_in[7];
  float* out = (float*)d_out;

  const int* src = ei;            // edge_index[0, :]
  const int* dst = ei + N_EDGES;  // edge_index[1, :]

  float* sum = (float*)d_ws;                    // N*F
  float* cnt = sum + (size_t)N_NODES * F;       // N
  float* h = cnt + N_NODES;                     // N*F

  int tileBlocks = (N_NODES + 63) / 64;  // 4 wave-tiles of 16 rows per block

  // ---- layer 1 aggregation
  zero_f32<<<2048, 256, 0, stream>>>(sum, N_NODES * F + N_NODES);  // sum+cnt contiguous
  degree_count<<<(N_EDGES + 255) / 256, 256, 0, stream>>>(dst, cnt, N_EDGES);
  scatter_add<<<(N_EDGES * F + 255) / 256, 256, 0, stream>>>(x, src, dst, sum, N_EDGES);
  sage_layer1_wmma<<<tileBlocks, 128, 0, stream>>>(x, sum, cnt, W1l, W1r, b1, h);

  // ---- layer 2 aggregation (same graph -> reuse cnt)
  zero_f32<<<2048, 256, 0, stream>>>(sum, N_NODES * F);
  scatter_add<<<(N_EDGES * F + 255) / 256, 256, 0, stream>>>(h, src, dst, sum, N_EDGES);
  sage_layer2_wmma<<<tileBlocks, 128, 0, stream>>>(h, sum, cnt, W2l, b2, W2r, out);

  // ---- second tuple element: pass edge_index through to the output tail
  long long tailElems = (long long)out_size - (long long)N_NODES * NCLS;
  if (tailElems > 0) {
    size_t bytes = (size_t)tailElems * sizeof(float);
    size_t avail = (size_t)in_sizes[1] * sizeof(int);
    if (bytes > avail) bytes = avail;
    hipMemcpyAsync(out + (size_t)N_NODES * NCLS, d_in[1], bytes,
                   hipMemcpyDeviceToDevice, stream);
  }
}